// COVHOPF_86148454023411
// MI455X (gfx1250) — compile-verified
//
#include <hip/hip_runtime.h>

// Hopf CSD covariance model for MI455X (gfx1250).
//   - kernel 1: zero the 64x64 channel-space accumulator E (in d_ws)
//   - kernel 2: w_n = g * exp(wll).*sc / rowsum  (precompute, d_ws)
//   - kernel 3: one workgroup (1024 thr = 32 waves) per frequency:
//       build M in LDS (200x200 complex = 312.6KB -> fits 320KB WGP LDS),
//       in-place Gauss-Jordan inverse (no pivoting; diagonally dominant),
//       G = lm @ H via V_WMMA_F32_16X16X4_F32 (fp32-exact, A-fragment reuse),
//       E += Gr Gr^T + Gi Gi^T via WMMA, f32 atomics into global E.
//   - kernel 4: out = cy0 * std_in^2 * df * E
//
// All launches on `stream`; only d_ws scratch; deterministic.

#define NNODE   200
#define MPAD    201          // LDS row stride for M planes
#define NCHAN   64
#define NFREQ   512
#define GSTRIDE 208          // padded col stride for the 64x200 G planes
#define NTHREADS 1024

typedef __attribute__((ext_vector_type(2))) float v2f;
typedef __attribute__((ext_vector_type(8))) float v8f;

#define SMEM_FLOATS (2 * NNODE * MPAD + 4 * NNODE + 2)
#define SMEM_BYTES  (SMEM_FLOATS * 4)

__global__ void COVHOPF_zero_kernel(float* __restrict__ E) {
    int i = blockIdx.x * blockDim.x + threadIdx.x;
    if (i < NCHAN * NCHAN) E[i] = 0.0f;
}

// w_n = g * (exp(wll) .* sc) / (rowsum + 1e-8)
__global__ void COVHOPF_wn_kernel(const float* __restrict__ sc,
                                  const float* __restrict__ wll,
                                  const float* __restrict__ g,
                                  float* __restrict__ wn) {
    __shared__ float red[256];
    const int row = blockIdx.x;
    float s = 0.0f;
    for (int j = threadIdx.x; j < NNODE; j += 256)
        s += __expf(wll[row * NNODE + j]) * sc[row * NNODE + j];
    red[threadIdx.x] = s;
    __syncthreads();
    for (int off = 128; off > 0; off >>= 1) {
        if (threadIdx.x < off) red[threadIdx.x] += red[threadIdx.x + off];
        __syncthreads();
    }
    const float scale = g[0] / (red[0] + 1e-8f);
    for (int j = threadIdx.x; j < NNODE; j += 256)
        wn[row * NNODE + j] = __expf(wll[row * NNODE + j]) * sc[row * NNODE + j] * scale;
}

__global__ void __launch_bounds__(NTHREADS, 1)
COVHOPF_freq_kernel(const float* __restrict__ dist,
                    const float* __restrict__ freqs,
                    const float* __restrict__ lm,
                    const float* __restrict__ wn,
                    const float* __restrict__ a_p,
                    const float* __restrict__ omega_p,
                    const float* __restrict__ vd_p,
                    float* __restrict__ E) {
    extern __shared__ float smem[];
    float* Mr  = smem;                    // NNODE*MPAD
    float* Mi  = Mr + NNODE * MPAD;       // NNODE*MPAD
    float* rkr = Mi + NNODE * MPAD;       // NNODE  (scaled row-k snapshot u, real)
    float* rki = rkr + NNODE;             // NNODE
    float* fcr = rki + NNODE;             // NNODE  (col-k snapshot, real)
    float* fci = fcr + NNODE;             // NNODE
    float* pv  = fci + NNODE;             // 2      (pivot reciprocal broadcast)

    const int tid  = threadIdx.x;
    const int lane = tid & 31;
    const int wv   = tid >> 5;

    const float om     = 6.2831853071795864f * freqs[blockIdx.x];
    const float aval   = a_p[0];
    const float w0     = omega_p[0];
    const float inv_vd = 1.0f / vd_p[0];

    // ---- build M = (i*om - a - i*omega0) I - w_n .* exp(-i*om*tau) ----
    for (int idx = tid; idx < NNODE * NNODE; idx += NTHREADS) {
        const int i = idx / NNODE, j = idx - i * NNODE;
        const float w  = wn[idx];
        const float th = om * dist[idx] * inv_vd;
        float sn, cs;
        __sincosf(th, &sn, &cs);
        float mr = -w * cs;
        float mi =  w * sn;
        if (i == j) { mr -= aval; mi += om - w0; }
        Mr[i * MPAD + j] = mr;
        Mi[i * MPAD + j] = mi;
    }
    __syncthreads();

    // ---- in-place complex Gauss-Jordan inverse (no pivoting) ----
    // Fixed mapping: 5 threads per row x 40-column strips (1000 active threads);
    // no div/mod inside the k-loop.
    const int irow = tid / 5;             // 0..204
    const int j0   = (tid - irow * 5) * 40;
    const bool act = (irow < NNODE);

    for (int k = 0; k < NNODE; ++k) {
        // Phase A: snapshot col k; snapshot row k pre-scaled by pivot recip.
        if (tid < NNODE) {
            const float dr = Mr[k * MPAD + k], di = Mi[k * MPAD + k];
            const float inv = 1.0f / (dr * dr + di * di);
            const float pr = dr * inv, pi = -di * inv;     // p = 1/M[k][k]
            const float ar = Mr[k * MPAD + tid], ai = Mi[k * MPAD + tid];
            rkr[tid] = ar * pr - ai * pi;                  // u[j] = M[k][j]*p
            rki[tid] = ar * pi + ai * pr;
            fcr[tid] = Mr[tid * MPAD + k];                 // f[i] = M[i][k]
            fci[tid] = Mi[tid * MPAD + k];
            if (tid == 0) { pv[0] = pr; pv[1] = pi; }
        }
        __syncthreads();

        // Phase B: rank-1 update / row-k write / fixups.
        if (act) {
            const float pr = pv[0], pi = pv[1];
            if (irow == k) {
                for (int j = j0; j < j0 + 40; ++j) {
                    Mr[k * MPAD + j] = rkr[j];
                    Mi[k * MPAD + j] = rki[j];
                }
                if (j0 <= k && k < j0 + 40) {
                    Mr[k * MPAD + k] = pr;
                    Mi[k * MPAD + k] = pi;
                }
            } else {
                const float fr = fcr[irow], fim = fci[irow];
                for (int j = j0; j < j0 + 40; ++j) {
                    const float ur = rkr[j], ui = rki[j];
                    Mr[irow * MPAD + j] -= fr * ur - fim * ui;
                    Mi[irow * MPAD + j] -= fr * ui + fim * ur;
                }
                if (j0 <= k && k < j0 + 40) {              // exact col-k entry
                    Mr[irow * MPAD + k] = -(fr * pr - fim * pi);
                    Mi[irow * MPAD + k] = -(fr * pi + fim * pr);
                }
            }
        }
        __syncthreads();
    }
    // Mr/Mi now hold H = M^{-1} (real/imag planes).

    // ---- G = lm @ H (64x200, 2 planes) via WMMA f32 16x16x4 ----
    // 8 row-strips (plane x mt); 4 waves per strip; each wave owns 4 nt-tiles
    // so one A fragment feeds 4 WMMAs per k-step. Tiles past nt=12 are
    // duplicates whose results are discarded (keeps control flow uniform).
    const int m    = lane & 15;
    const int kb   = (lane >> 4) * 2;      // lanes 0-15: K=0,1 ; 16-31: K=2,3
    const int grp  = wv >> 2;              // 0..7
    const int sub  = wv & 3;               // 0..3
    const int plane = grp >> 2;            // 0..1
    const int mt    = grp & 3;             // 0..3
    const float* Hp = plane ? Mi : Mr;

    int colt[4];
#pragma unroll
    for (int t = 0; t < 4; ++t) {
        int nt = sub + 4 * t;
        if (nt > 12) nt = 12;              // dummy duplicate, discarded later
        colt[t] = nt * 16 + m;
    }

    v8f acc[4] = {};
    for (int k0 = 0; k0 < NNODE; k0 += 4) {
        v2f A;
        A.x = lm[(mt * 16 + m) * NNODE + k0 + kb];
        A.y = lm[(mt * 16 + m) * NNODE + k0 + kb + 1];
#pragma unroll
        for (int t = 0; t < 4; ++t) {
            v2f B;                          // unconditional: pads never consumed
            B.x = Hp[(k0 + kb) * MPAD + colt[t]];
            B.y = Hp[(k0 + kb + 1) * MPAD + colt[t]];
            acc[t] = __builtin_amdgcn_wmma_f32_16x16x4_f32(
                false, A, false, B, (short)0, acc[t], false, false);
        }
    }
    __syncthreads();   // all reads of H done -> LDS reusable

    // ---- stage G (64 x GSTRIDE per plane) into LDS ----
    float* G = smem;   // Gr at 0, Gi at NCHAN*GSTRIDE (106,496B < M region)
    {
        const int rbase = mt * 16 + ((lane < 16) ? 0 : 8);
        float* Gp = G + plane * (NCHAN * GSTRIDE);
#pragma unroll
        for (int t = 0; t < 4; ++t) {
            const int nt = sub + 4 * t;
            if (nt < 13) {
#pragma unroll
                for (int r = 0; r < 8; ++r)
                    Gp[(rbase + r) * GSTRIDE + (nt * 16 + m)] = acc[t][r];
            }
        }
    }
    __syncthreads();

    // ---- E += Gr Gr^T + Gi Gi^T via WMMA; 16 tiles x 2 K-halves of 100 ----
    {
        const int tile = wv & 15;
        const int half = wv >> 4;
        const int it = tile >> 2, jt = tile & 3;
        v8f c = {};
        const int ks = half * 100, kend = ks + 100;   // 200 = 50x4, no pads read
        for (int k0 = ks; k0 < kend; k0 += 4) {
#pragma unroll
            for (int p = 0; p < 2; ++p) {
                const float* Gp = G + p * (NCHAN * GSTRIDE);
                v2f A, B;
                A.x = Gp[(it * 16 + m) * GSTRIDE + k0 + kb];
                A.y = Gp[(it * 16 + m) * GSTRIDE + k0 + kb + 1];
                B.x = Gp[(jt * 16 + m) * GSTRIDE + k0 + kb];  // B[kk][n]=G[jt*16+n][k]
                B.y = Gp[(jt * 16 + m) * GSTRIDE + k0 + kb + 1];
                c = __builtin_amdgcn_wmma_f32_16x16x4_f32(
                    false, A, false, B, (short)0, c, false, false);
            }
        }
        const int colw  = jt * 16 + m;
        const int rbase = it * 16 + ((lane < 16) ? 0 : 8);
#pragma unroll
        for (int r = 0; r < 8; ++r)
            unsafeAtomicAdd(&E[(rbase + r) * NCHAN + colw], c[r]);
    }
}

__global__ void COVHOPF_final_kernel(const float* __restrict__ E,
                                     const float* __restrict__ freqs,
                                     const float* __restrict__ std_p,
                                     const float* __restrict__ cy0_p,
                                     float* __restrict__ out) {
    int i = blockIdx.x * blockDim.x + threadIdx.x;
    if (i < NCHAN * NCHAN) {
        const float df = freqs[1] - freqs[0];
        const float s  = cy0_p[0] * std_p[0] * std_p[0] * df;
        out[i] = E[i] * s;
    }
}

extern "C" void kernel_launch(void* const* d_in, const int* in_sizes, int n_in,
                              void* d_out, int out_size, void* d_ws, size_t ws_size,
                              hipStream_t stream) {
    const float* sc     = (const float*)d_in[0];
    const float* dist   = (const float*)d_in[1];
    const float* freqs  = (const float*)d_in[2];
    const float* lm     = (const float*)d_in[3];
    const float* wll    = (const float*)d_in[4];
    const float* a      = (const float*)d_in[5];
    const float* omega  = (const float*)d_in[6];
    const float* g      = (const float*)d_in[7];
    const float* std_in = (const float*)d_in[8];
    const float* v_d    = (const float*)d_in[9];
    const float* cy0    = (const float*)d_in[10];
    float* out = (float*)d_out;

    float* wn = (float*)d_ws;              // NNODE*NNODE
    float* E  = wn + NNODE * NNODE;        // NCHAN*NCHAN accumulator

    // Allow ~317KB dynamic LDS (fits the 320KB WGP LDS of CDNA5).
    (void)hipFuncSetAttribute((const void*)COVHOPF_freq_kernel,
                              hipFuncAttributeMaxDynamicSharedMemorySize, SMEM_BYTES);

    COVHOPF_zero_kernel<<<(NCHAN * NCHAN + 255) / 256, 256, 0, stream>>>(E);
    COVHOPF_wn_kernel<<<NNODE, 256, 0, stream>>>(sc, wll, g, wn);
    COVHOPF_freq_kernel<<<NFREQ, NTHREADS, SMEM_BYTES, stream>>>(
        dist, freqs, lm, wn, a, omega, v_d, E);
    COVHOPF_final_kernel<<<(NCHAN * NCHAN + 255) / 256, 256, 0, stream>>>(
        E, freqs, std_in, cy0, out);
}